// Anki_31585189494809
// MI455X (gfx1250) — compile-verified
//
#include <hip/hip_runtime.h>
#include <hip/hip_bf16.h>

// Anki scheduler scan: 64 sequential steps over 262144 independent columns.
// Memory-bound streaming (~258 MiB @ 23.3 TB/s ~ 11 us floor). One thread
// carries 2 columns of (ivl, ease) in VGPRs; all global traffic is NT-hinted
// b128. Every data-dependent choice is an empty-diamond select -> v_cndmask
// (no divergent branches, no exec-mask SALU churn).

#define SEQ_LEN 64
#define BATCH   262144

typedef float v4f __attribute__((ext_vector_type(4)));

// Branchless select: after inlining, arms are plain SSA values, so the
// empty diamond always folds to an IR `select` -> single v_cndmask_b32.
__device__ __forceinline__ float fsel(bool c, float a, float b) {
    return c ? a : b;
}

__device__ __forceinline__ void anki_step(
    float delta_t, float rating,
    float w3, float w4, float w5, float w6, float c3,
    float& ivl, float& ease)
{
#pragma clang fp contract(off)          // match XLA's un-contracted rounding
    const float days_late = delta_t - ivl;
    const float elapsed   = ivl + days_late;       // == (ivl + days_late) in ref

    const bool r1 = (rating == 1.0f);
    const bool r2 = (rating == 2.0f);
    const bool r4 = (rating == 4.0f);

    const float ee  = elapsed * ease;              // reused 2x
    const float me  = fmaxf(ee, ivl);
    const float iw4 = ivl * w4;                    // reused 2x

    // early = r2 ? max(elapsed*w4, ivl*w4/2) : r4 ? me : me*c3
    const float e_r2  = fmaxf(elapsed * w4, iw4 * 0.5f);
    const float e_oth = me * c3;
    const float early = fsel(r2, e_r2, fsel(r4, me, e_oth));

    // nonearly = r2 ? ivl*w4 : r4 ? (ivl+days_late/2)*ease : ((ivl+days_late)*ease)*w3
    const float n_r4  = (ivl + days_late * 0.5f) * ease;
    const float n_oth = ee * w3;
    const float nonearly = fsel(r2, iw4, fsel(r4, n_r4, n_oth));

    const float sel  = fsel(days_late < 0.0f, early, nonearly);
    const float ui   = fsel(r1, ivl * w5, sel * w6);

    // upd_ease = ease + {r1:-0.2, r2:-0.15, r4:+0.15, else 0}; bit-identical
    // to the reference (ease in [1.3,5.5] after step 0, so +0.0f is identity).
    const float ed = fsel(r1, -0.2f, fsel(r2, -0.15f, fsel(r4, 0.15f, 0.0f)));
    const float ue = ease + ed;

    ease = fminf(fmaxf(ue, 1.3f), 5.5f);
    const float x  = ui - 1.0f;
    const float lr = fsel(x >= 0.0f, x, x * 0.01f);     // leaky_relu(ui-1)
    ivl = fminf(fmaxf(fmaxf(lr + 1.0f, ui), 0.01f), 36500.0f);
}

// new_ivl transform applied to the t==0 init interval (scalar, once per thread)
__device__ __forceinline__ float init_transform(float v)
{
#pragma clang fp contract(off)
    const float x  = v - 1.0f;
    const float lr = fsel(x >= 0.0f, x, x * 0.01f);
    return fminf(fmaxf(fmaxf(lr + 1.0f, v), 0.01f), 36500.0f);
}

__global__ __launch_bounds__(256) void anki_scan_kernel(
    const float* __restrict__ in,   // (SEQ_LEN, BATCH, 2)
    const float* __restrict__ w,    // (7,)
    float*       __restrict__ out)  // (SEQ_LEN, BATCH, 2) then (BATCH, 2)
{
    const int pair = blockIdx.x * blockDim.x + threadIdx.x;   // 2 columns/thread
    if (pair >= BATCH / 2) return;

    // Uniform address + constant indices -> scalar loads / scalar float ops.
    const float w0 = w[0], w1 = w[1], w2 = w[2], w3 = w[3],
                w4 = w[4], w5 = w[5], w6 = w[6];
    const float c3  = w3 - (w3 - 1.0f) * 0.5f;
    const float niA = init_transform(w0);             // rating < 4 at t==0
    const float niB = init_transform(w1);             // rating == 4 at t==0
    const float ne0 = fminf(fmaxf(w2, 1.3f), 5.5f);   // init ease, clamped

    const size_t base      = (size_t)pair * 4u;       // floats
    const size_t rowstride = (size_t)BATCH * 2u;      // floats per time-step

    // ---- t = 0 peeled: state depends only on rating ----
    const v4f X0 = __builtin_nontemporal_load(reinterpret_cast<const v4f*>(in + base));
    float ivl0  = fsel(X0.y < 4.0f, niA, niB);
    float ease0 = ne0;
    float ivl1  = fsel(X0.w < 4.0f, niA, niB);
    float ease1 = ne0;
    {
        v4f Y; Y.x = ivl0; Y.y = ease0; Y.z = ivl1; Y.w = ease1;
        __builtin_nontemporal_store(Y, reinterpret_cast<v4f*>(out + base));
    }

    // ---- steady state: t = 1 .. 63 ----
    #pragma clang loop unroll_count(2)
    for (int t = 1; t < SEQ_LEN; ++t) {
        const size_t off = (size_t)t * rowstride + base;

        // Speculative prefetch ~8 rows (16 MB) ahead; dropped if unmapped.
        __builtin_prefetch(in + off + 8u * rowstride, 0, 0);

        const v4f X = __builtin_nontemporal_load(
            reinterpret_cast<const v4f*>(in + off));

        anki_step(X.x, X.y, w3, w4, w5, w6, c3, ivl0, ease0);
        anki_step(X.z, X.w, w3, w4, w5, w6, c3, ivl1, ease1);

        v4f Y;
        Y.x = ivl0; Y.y = ease0; Y.z = ivl1; Y.w = ease1;
        __builtin_nontemporal_store(Y, reinterpret_cast<v4f*>(out + off));
    }

    // final_state appended after the 64 output rows
    v4f F;
    F.x = ivl0; F.y = ease0; F.z = ivl1; F.w = ease1;
    __builtin_nontemporal_store(
        F, reinterpret_cast<v4f*>(out + (size_t)SEQ_LEN * rowstride + base));
}

extern "C" void kernel_launch(void* const* d_in, const int* in_sizes, int n_in,
                              void* d_out, int out_size, void* d_ws, size_t ws_size,
                              hipStream_t stream) {
    (void)in_sizes; (void)n_in; (void)out_size; (void)d_ws; (void)ws_size;
    const float* in  = (const float*)d_in[0];   // (64, 262144, 2) f32
    const float* w   = (const float*)d_in[1];   // (7,) f32
    float*       out = (float*)d_out;

    const int threads = 256;
    const int pairs   = BATCH / 2;              // 131072
    const int blocks  = (pairs + threads - 1) / threads;  // 512
    anki_scan_kernel<<<blocks, threads, 0, stream>>>(in, w, out);
}